// DiagonalDense_82197084111240
// MI455X (gfx1250) — compile-verified
//
#include <hip/hip_runtime.h>

// out[r, c] = x[r, c] * diag[c]   with x: 16384 x 2048 fp32, diag: 2048 fp32.
// Memory-bound: 256 MiB HBM traffic, ~11.5 us floor at 23.3 TB/s. WMMA would
// be a strict pessimization (0.125 FLOP/byte); the chip rewards wide NT
// streaming here, not matrix cores.
//
// Layout: block tile = 2048 float4s (exactly 4 rows; row = 512 float4s).
// Thread t at unroll step u touches flat float4 index
//   i = blockIdx*2048 + t + u*256  =>  col4 = t + (u&1)*256.
// So each thread needs only TWO diag float4s, held in registers.
//
// All indices are 32-bit: total4 = 8,388,608 (byte offset 128 MiB < 2^31),
// so the compiler can use SGPR-base + 32-bit VGPR offset (scale_offset GVS
// addressing) for every stream — no 64-bit per-lane address math.

typedef float v4f __attribute__((ext_vector_type(4)));

#define TPB     256
#define UNROLL  8
#define TILE4   (TPB * UNROLL)   // 2048 float4s = 32 KB per block

__global__ __launch_bounds__(TPB) void DiagonalDense_scale_kernel(
    const v4f* __restrict__ x4,
    const v4f* __restrict__ diag4,
    v4f* __restrict__ out4,
    int total4)
{
    const int tid  = (int)threadIdx.x;
    const int base = (int)blockIdx.x * TILE4 + tid;

    // The only two column groups this thread ever sees (RT hint: diag stays
    // hot in WGP$/L2 across all 4096 blocks).
    const v4f d0 = diag4[tid];        // even unroll steps
    const v4f d1 = diag4[tid + TPB];  // odd  unroll steps

    if (base + (UNROLL - 1) * TPB < total4) {
        // Fast path: 8 independent NT b128 loads issued before any use
        // (deep MLP per wave), then 8 v_pk_mul_f32 pairs + 8 NT b128 stores.
        v4f v[UNROLL];
        #pragma unroll
        for (int u = 0; u < UNROLL; ++u)
            v[u] = __builtin_nontemporal_load(&x4[base + u * TPB]);

        #pragma unroll
        for (int u = 0; u < UNROLL; ++u)
            __builtin_nontemporal_store(v[u] * ((u & 1) ? d1 : d0),
                                        &out4[base + u * TPB]);
    } else {
        // Tail path (unused for the reference shape, kept for correctness).
        #pragma unroll
        for (int u = 0; u < UNROLL; ++u) {
            const int i = base + u * TPB;
            if (i < total4) {
                v4f v = __builtin_nontemporal_load(&x4[i]);
                __builtin_nontemporal_store(v * ((u & 1) ? d1 : d0), &out4[i]);
            }
        }
    }
}

extern "C" void kernel_launch(void* const* d_in, const int* in_sizes, int n_in,
                              void* d_out, int out_size, void* d_ws, size_t ws_size,
                              hipStream_t stream) {
    const float* x    = (const float*)d_in[0];       // 16384 * 2048 fp32
    const float* diag = (const float*)d_in[1];       // 2048 fp32
    float* out        = (float*)d_out;

    const int total  = in_sizes[0];                  // 33,554,432 elements
    const int total4 = total / 4;                    // 8,388,608 float4s

    // 4096 blocks x 256 threads (8 wave32s/block); each block streams a
    // contiguous 32 KB tile. 8 outstanding b128 loads per thread x many
    // waves per WGP saturates 23.3 TB/s.
    const int grid = (total4 + TILE4 - 1) / TILE4;

    DiagonalDense_scale_kernel<<<grid, TPB, 0, stream>>>(
        (const v4f*)x, (const v4f*)diag, (v4f*)out, total4);
}